// CTPNLoss_30210799960881
// MI455X (gfx1250) — compile-verified
//
#include <hip/hip_runtime.h>
#include <hip/hip_bf16.h>
#include <math.h>

// ---------- CDNA5 WMMA types ----------
typedef float v2f __attribute__((ext_vector_type(2)));
typedef float v8f __attribute__((ext_vector_type(8)));

__device__ __forceinline__ float smooth_l1f(float d) {
  float a = fabsf(d);
  return (a < 1.0f) ? 0.5f * a * a : a - 0.5f;
}

// jnp.logaddexp(a,b) = max + log1p(exp(min - max))
__device__ __forceinline__ float logaddexpf_(float a, float b) {
  float m = fmaxf(a, b);
  float n = fminf(a, b);
  return m + log1pf(expf(n - m));
}

__device__ __forceinline__ float read_lane_f(float v, unsigned l) {
  return __builtin_bit_cast(
      float, __builtin_amdgcn_readlane(__builtin_bit_cast(int, v), l));
}

// Sum a value across all 32 lanes of a wave32 using the matrix core.
// A (16x4 f32): lane L<16 holds A[L][0]=v, A[L][1]=0; lane L>=16 holds
// A[L-16][2]=v, A[L-16][3]=0.  B = all-ones 4x16.  Then
//   D[m][n] = v[m] + v[m+16]   (replicated over n)
// Per-lane sum of the 8 D VGPRs = rows 0..7 (lanes 0-15) or rows 8..15
// (lanes 16-31); combine the two halves with v_readlane.
// Requires EXEC == all ones (guaranteed: blockDim.x == 32, no early exit).
__device__ __forceinline__ float wave_sum32(float v) {
  v2f a; a[0] = v;    a[1] = 0.0f;
  v2f b; b[0] = 1.0f; b[1] = 1.0f;
  v8f c = {};
  c = __builtin_amdgcn_wmma_f32_16x16x4_f32(false, a, false, b, (short)0, c,
                                            false, false);
  float s = ((c[0] + c[1]) + (c[2] + c[3])) + ((c[4] + c[5]) + (c[6] + c[7]));
  return read_lane_f(s, 0) + read_lane_f(s, 16);
}

// x: (1, 512, 1024, 50) fp32;  K=10 anchors, C = 5*K = 50 channels.
// Single wave handles all 64 pos + 64 neg anchors (2+2 per lane).
__global__ __launch_bounds__(32) void ctpn_loss_kernel(
    const float* __restrict__ X,
    const float* __restrict__ v_targets,   // [64,2]
    const float* __restrict__ o_targets,   // [64]
    const int* __restrict__ pos_y, const int* __restrict__ pos_x,
    const int* __restrict__ pos_z,
    const int* __restrict__ neg_y, const int* __restrict__ neg_x,
    const int* __restrict__ neg_z,
    const unsigned char* __restrict__ o_mask,  // [64] bool bytes
    float* __restrict__ out) {
  const int lane = threadIdx.x;  // 0..31, wave32
  const int W = 1024, C = 50;

  float s_ce = 0.0f, s_lv = 0.0f, s_lo = 0.0f, s_cnt = 0.0f;

#pragma unroll
  for (int t = 0; t < 2; ++t) {
    const int i = lane + 32 * t;

    // ---- positive anchor i: CE (label 1), vertical reg, side refinement ----
    {
      const int y = pos_y[i], xx = pos_x[i], z = pos_z[i];
      const float* p = X + ((size_t)y * W + (size_t)xx) * C;
      const float p0 = p[20 + 2 * z];
      const float p1 = p[21 + 2 * z];
      s_ce += logaddexpf_(p0, p1) - p1;

      const float v0 = p[2 * z];
      const float v1 = p[2 * z + 1];
      s_lv += 0.5f * (smooth_l1f(v0 - v_targets[2 * i]) +
                      smooth_l1f(v1 - v_targets[2 * i + 1]));

      const float op  = p[40 + z];
      const float lo  = smooth_l1f(op - o_targets[i]);
      if (o_mask[i] != 0) {  // reconverges before any WMMA
        s_lo  += lo;
        s_cnt += 1.0f;
      }
    }

    // ---- negative anchor i: CE (label 0) ----
    {
      const int y = neg_y[i], xx = neg_x[i], z = neg_z[i];
      const float* p = X + ((size_t)y * W + (size_t)xx) * C;
      const float n0 = p[20 + 2 * z];
      const float n1 = p[21 + 2 * z];
      s_ce += logaddexpf_(n0, n1) - n0;
    }
  }

  // ---- 32-lane reductions on the matrix core (EXEC all-ones here) ----
  const float ce_tot = wave_sum32(s_ce);
  const float lv_tot = wave_sum32(s_lv);
  const float lo_tot = wave_sum32(s_lo);
  const float n_o    = wave_sum32(s_cnt);  // >= 1 by construction

  if (lane == 0) {
    // loss_cls averaged over Np+Nn = 128; both reg terms divided by n_o
    out[0] = ce_tot * (1.0f / 128.0f) + lv_tot / n_o + lo_tot / n_o;
  }
}

extern "C" void kernel_launch(void* const* d_in, const int* in_sizes, int n_in,
                              void* d_out, int out_size, void* d_ws,
                              size_t ws_size, hipStream_t stream) {
  (void)in_sizes; (void)n_in; (void)out_size; (void)d_ws; (void)ws_size;
  const float* X        = (const float*)d_in[0];
  const float* v_tgt    = (const float*)d_in[1];
  const float* o_tgt    = (const float*)d_in[2];
  const int*   pos_y    = (const int*)d_in[3];
  const int*   pos_x    = (const int*)d_in[4];
  const int*   pos_z    = (const int*)d_in[5];
  const int*   neg_y    = (const int*)d_in[6];
  const int*   neg_x    = (const int*)d_in[7];
  const int*   neg_z    = (const int*)d_in[8];
  const unsigned char* o_mask = (const unsigned char*)d_in[9];
  float* out = (float*)d_out;

  // Latency-bound problem: one kernel, one wave, no barriers.
  ctpn_loss_kernel<<<1, 32, 0, stream>>>(X, v_tgt, o_tgt, pos_y, pos_x, pos_z,
                                         neg_y, neg_x, neg_z, o_mask, out);
}